// MultiStepMSELoss_30794915512933
// MI455X (gfx1250) — compile-verified
//
#include <hip/hip_runtime.h>

typedef float v2f __attribute__((ext_vector_type(2)));
typedef float v4f __attribute__((ext_vector_type(4)));
typedef float v8f __attribute__((ext_vector_type(8)));

#define NBLK   1024
#define NTHR   256
#define NWAVES (NTHR / 32)
#define T_STEPS 50
#define D_FEAT  128

// Full-wave (32 lane) sum using V_WMMA_F32_16X16X4_F32.
// A-matrix (16x4 f32): lane m (m<16) supplies A[m,0..1] in 2 VGPRs,
// lane m+16 supplies A[m,2..3]. We set A = {acc, 0}, B = ones, C = 0:
//   D[m,n] = acc[m] + acc[m+16]  (replicated over n)
// Each lane holds D rows m = j + 8*(lane/16) in its 8 dest VGPRs, so a
// per-lane sum of the 8 VGPRs + one shfl_xor(16) yields the full wave sum.
// Requires EXEC == all ones (called with no divergence).
__device__ __forceinline__ float wave_sum_wmma(float acc) {
  v2f a; a[0] = acc;  a[1] = 0.0f;
  v2f b; b[0] = 1.0f; b[1] = 1.0f;
  v8f c = {0.f, 0.f, 0.f, 0.f, 0.f, 0.f, 0.f, 0.f};
  v8f d = __builtin_amdgcn_wmma_f32_16x16x4_f32(
      /*neg_a=*/false, a, /*neg_b=*/false, b,
      /*c_mod=*/(short)0, c, /*reuse_a=*/false, /*reuse_b=*/false);
  float s = ((d[0] + d[1]) + (d[2] + d[3])) + ((d[4] + d[5]) + (d[6] + d[7]));
  s += __shfl_xor(s, 16, 32);   // combine the two half-wave row groups
  return s;                      // valid in every lane
}

// Kernel 1: one wave processes one (b,t) row of 128 features per iteration.
// Lane l owns features [4l, 4l+4). Mask test is wave-uniform -> masked rows
// cost zero memory traffic (~2x saving for U[0,48] break_steps).
__global__ __launch_bounds__(NTHR)
void msel_partial_kernel(const float* __restrict__ yp,
                         const float* __restrict__ db,
                         const int*   __restrict__ bs,
                         const float* __restrict__ w,
                         float* __restrict__ partials,
                         int Btot) {
  const int  lane      = threadIdx.x & 31;
  const int  waveInBlk = threadIdx.x >> 5;
  const long waveId    = (long)blockIdx.x * NWAVES + waveInBlk;
  const long numWaves  = (long)gridDim.x * NWAVES;
  const long R         = (long)Btot * T_STEPS;

  const v4f w4 = *(const v4f*)(w + 4 * lane);   // weights resident in VGPRs

  float acc = 0.0f;
  for (long r = waveId; r < R; r += numWaves) {
    const long b   = r / T_STEPS;
    const int  t   = (int)(r - b * T_STEPS);
    const int  cnt = bs[b] + 1;                 // wave-uniform
    if (t < cnt) {
      const float inv  = 1.0f / (float)cnt;
      const long  base = r * D_FEAT + 4 * lane;
      // streamed once; 335MB > 192MB L2 -> non-temporal 128b loads
      v4f y  = __builtin_nontemporal_load((const v4f*)(yp + base));
      v4f dl = __builtin_nontemporal_load((const v4f*)(db + base));
      v4f df = y - dl;
      float p = fmaf(w4[0] * df[0], df[0],
                fmaf(w4[1] * df[1], df[1],
                fmaf(w4[2] * df[2], df[2],
                     w4[3] * df[3] * df[3])));
      acc = fmaf(inv, p, acc);
    }
  }

  // Wave reduction via CDNA5 WMMA (all lanes active here), then block reduce.
  float wsum = wave_sum_wmma(acc);
  __shared__ float lds[NWAVES];
  if (lane == 0) lds[waveInBlk] = wsum;
  __syncthreads();
  if (threadIdx.x == 0) {
    float s = 0.0f;
    #pragma unroll
    for (int i = 0; i < NWAVES; ++i) s += lds[i];
    partials[blockIdx.x] = s;                   // deterministic: no atomics
  }
}

// Kernel 2: deterministic reduction of the NBLK block partials.
__global__ __launch_bounds__(NTHR)
void msel_finalize_kernel(const float* __restrict__ partials,
                          float* __restrict__ out, float scale) {
  __shared__ float lds[NTHR];
  float s = 0.0f;
  for (int i = threadIdx.x; i < NBLK; i += NTHR) s += partials[i];
  lds[threadIdx.x] = s;
  __syncthreads();
  for (int off = NTHR / 2; off > 0; off >>= 1) {
    if (threadIdx.x < off) lds[threadIdx.x] += lds[threadIdx.x + off];
    __syncthreads();
  }
  if (threadIdx.x == 0) out[0] = lds[0] * scale;
}

extern "C" void kernel_launch(void* const* d_in, const int* in_sizes, int n_in,
                              void* d_out, int out_size, void* d_ws, size_t ws_size,
                              hipStream_t stream) {
  const float* yp = (const float*)d_in[0];   // y_pred        [B,T,D] f32
  const float* db = (const float*)d_in[1];   // delta_batch   [B,T,D] f32
  const int*   bs = (const int*)  d_in[2];   // break_steps   [B]     i32
  const float* w  = (const float*)d_in[3];   // norm_loss_wghts [D]   f32
  float* out      = (float*)d_out;
  float* partials = (float*)d_ws;            // NBLK floats of scratch

  const int Btot = in_sizes[2];
  msel_partial_kernel<<<NBLK, NTHR, 0, stream>>>(yp, db, bs, w, partials, Btot);

  const float scale = 1.0f / ((float)Btot * (float)D_FEAT);
  msel_finalize_kernel<<<1, NTHR, 0, stream>>>(partials, out, scale);
}